// LocalAgg_52536039964748
// MI455X (gfx1250) — compile-verified
//
#include <hip/hip_runtime.h>
#include <hip/hip_bf16.h>

typedef __attribute__((ext_vector_type(2))) float v2f;
typedef __attribute__((ext_vector_type(8))) float v8f;

#define BATCH 4
#define NPTS  4096
#define CIN   64
#define CE    128      // 2*C edge feature dim
#define OUTC  128
#define KNN   16
#define MTOT  (BATCH*NPTS*KNN)   // 262144 rows
#define TS    132      // padded LDS row stride for activation tile
#define WSD   130      // padded LDS row stride for weights

// ---------------------------------------------------------------------------
// KNN: one thread per query point; tile candidate xyz through LDS; register
// insertion-sorted top-16 (strict < keeps lowest index on ties, like top_k).
// ---------------------------------------------------------------------------
__global__ __launch_bounds__(256) void knn_kernel(const float* __restrict__ xyz,
                                                  int* __restrict__ idxbuf) {
  __shared__ float sx[256], sy[256], sz[256];
  const int b = blockIdx.x >> 4;
  const int n = ((blockIdx.x & 15) << 8) + threadIdx.x;
  const float* base = xyz + (size_t)b * NPTS * 3;
  const float qx = base[n * 3 + 0];
  const float qy = base[n * 3 + 1];
  const float qz = base[n * 3 + 2];
  float bd[KNN];
  int   bi[KNN];
#pragma unroll
  for (int s = 0; s < KNN; ++s) { bd[s] = 3.4e38f; bi[s] = 0; }

  for (int t = 0; t < NPTS; t += 256) {
    __syncthreads();
    const int m0 = t + threadIdx.x;
    sx[threadIdx.x] = base[m0 * 3 + 0];
    sy[threadIdx.x] = base[m0 * 3 + 1];
    sz[threadIdx.x] = base[m0 * 3 + 2];
    __syncthreads();
    for (int mm = 0; mm < 256; ++mm) {
      const float dx = qx - sx[mm];
      const float dy = qy - sy[mm];
      const float dz = qz - sz[mm];
      const float d2 = dx * dx + dy * dy + dz * dz;
      if (d2 < bd[KNN - 1]) {
        bd[KNN - 1] = d2;
        bi[KNN - 1] = t + mm;
#pragma unroll
        for (int s = KNN - 1; s > 0; --s) {
          if (bd[s] < bd[s - 1]) {
            float td = bd[s]; bd[s] = bd[s - 1]; bd[s - 1] = td;
            int   ti = bi[s]; bi[s] = bi[s - 1]; bi[s - 1] = ti;
          }
        }
      }
    }
  }
  int* op = idxbuf + (size_t)(b * NPTS + n) * KNN;
#pragma unroll
  for (int s = 0; s < KNN; ++s) op[s] = bi[s];
}

// ---------------------------------------------------------------------------
// Fused gather + edge + GEMM1 (+ BN1/ReLU + GEMM2) (+ BN2/ReLU + k-max).
// phase 1: accumulate per-channel sum/sumsq of h1     (gsum/gsq = stats1)
// phase 2: BN1+ReLU, GEMM2, accumulate stats of h2    (gsum/gsq = stats2)
// phase 3: BN1+ReLU, GEMM2, BN2+ReLU, max over k -> out
// 256 threads = 8 waves; wave w owns global rows [blk*128 + 16w, +16) which is
// exactly one point's 16 neighbors (kk = row & 15).
// ---------------------------------------------------------------------------
__global__ __launch_bounds__(256) void fused_kernel(
    const float* __restrict__ feats, const int* __restrict__ idxbuf,
    const float* __restrict__ w1, const float* __restrict__ w2,
    const float* __restrict__ pa1, const float* __restrict__ pb1,
    const float* __restrict__ pa2, const float* __restrict__ pb2,
    float* __restrict__ gsum, float* __restrict__ gsq,
    float* __restrict__ out, int phase) {
  __shared__ float sW1[CE * WSD];     // sW1[c*WSD + o] = w1[o][c]
  __shared__ float sW2[OUTC * WSD];   // sW2[c*WSD + o] = w2[o][c]
  __shared__ float sT[128 * TS];      // edge tile, later bn1(h1) tile
  __shared__ float sSum[OUTC], sSq[OUTC];

  const int tid = threadIdx.x;
  const int wgStart = blockIdx.x * 128;

  // Stage weights (transposed) into LDS.
  for (int jj = tid; jj < OUTC * CE; jj += 256) {
    const int o = jj >> 7;
    const int c = jj & 127;
    sW1[c * WSD + o] = w1[jj];
    if (phase >= 2) sW2[c * WSD + o] = w2[jj];
  }

  // Build edge tile: 2 threads per row, 64 columns each.
  {
    const int r_local = tid >> 1;
    const int chalf   = tid & 1;
    const int r = wgStart + r_local;
    const int p = r >> 4;            // b*N + n
    const int b = p >> 12;
    const int j = idxbuf[r];         // within-batch neighbor index
    const float* ctr = feats + (size_t)p * CIN;
    const float* nb  = feats + ((size_t)((b << 12) + j)) * CIN;
    if (chalf == 0) {
#pragma unroll 8
      for (int c = 0; c < CIN; ++c) sT[r_local * TS + c] = ctr[c];
    } else {
#pragma unroll 8
      for (int c = 0; c < CIN; ++c) sT[r_local * TS + CIN + c] = nb[c] - ctr[c];
    }
  }
  __syncthreads();

  const int lane = tid & 31;
  const int wv   = tid >> 5;
  const int hh   = lane >> 4;   // wave half: K/M-row split per ISA layout
  const int l    = lane & 15;
  const int rowBase = wv * 16;

  const v8f vzero = {0.f, 0.f, 0.f, 0.f, 0.f, 0.f, 0.f, 0.f};
  v8f acc[8];
#pragma unroll
  for (int ct = 0; ct < 8; ++ct) acc[ct] = vzero;

  // GEMM1: h1 = edge @ w1^T  via chained V_WMMA_F32_16X16X4_F32.
  for (int k0 = 0; k0 < CE; k0 += 4) {
    v2f a;
    a.x = sT[(rowBase + l) * TS + k0 + 2 * hh];
    a.y = sT[(rowBase + l) * TS + k0 + 2 * hh + 1];
#pragma unroll
    for (int ct = 0; ct < 8; ++ct) {
      v2f bf;
      bf.x = sW1[(k0 + 2 * hh) * WSD + ct * 16 + l];
      bf.y = sW1[(k0 + 2 * hh + 1) * WSD + ct * 16 + l];
      acc[ct] = __builtin_amdgcn_wmma_f32_16x16x4_f32(
          false, a, false, bf, (short)0, acc[ct], false, false);
    }
  }

  if (phase == 1) {
    if (tid < OUTC) { sSum[tid] = 0.f; sSq[tid] = 0.f; }
    __syncthreads();
#pragma unroll
    for (int ct = 0; ct < 8; ++ct) {
      float s = 0.f, q = 0.f;
#pragma unroll
      for (int v = 0; v < 8; ++v) { const float x = acc[ct][v]; s += x; q += x * x; }
      atomicAdd(&sSum[ct * 16 + l], s);   // ds_add_f32
      atomicAdd(&sSq[ct * 16 + l], q);
    }
    __syncthreads();
    if (tid < OUTC) { atomicAdd(&gsum[tid], sSum[tid]); atomicAdd(&gsq[tid], sSq[tid]); }
    return;
  }

  // BN1 + ReLU, write back into sT (each wave owns its 16 rows).
#pragma unroll
  for (int ct = 0; ct < 8; ++ct) {
    const int ch = ct * 16 + l;
    const float sa = pa1[ch];
    const float sb = pb1[ch];
#pragma unroll
    for (int v = 0; v < 8; ++v) {
      const float x = fmaxf(acc[ct][v] * sa + sb, 0.f);
      sT[(rowBase + v + 8 * hh) * TS + ch] = x;
    }
  }
  __syncthreads();

  // GEMM2: h2 = h1' @ w2^T
#pragma unroll
  for (int ct = 0; ct < 8; ++ct) acc[ct] = vzero;
  for (int k0 = 0; k0 < OUTC; k0 += 4) {
    v2f a;
    a.x = sT[(rowBase + l) * TS + k0 + 2 * hh];
    a.y = sT[(rowBase + l) * TS + k0 + 2 * hh + 1];
#pragma unroll
    for (int ct = 0; ct < 8; ++ct) {
      v2f bf;
      bf.x = sW2[(k0 + 2 * hh) * WSD + ct * 16 + l];
      bf.y = sW2[(k0 + 2 * hh + 1) * WSD + ct * 16 + l];
      acc[ct] = __builtin_amdgcn_wmma_f32_16x16x4_f32(
          false, a, false, bf, (short)0, acc[ct], false, false);
    }
  }

  if (phase == 2) {
    if (tid < OUTC) { sSum[tid] = 0.f; sSq[tid] = 0.f; }
    __syncthreads();
#pragma unroll
    for (int ct = 0; ct < 8; ++ct) {
      float s = 0.f, q = 0.f;
#pragma unroll
      for (int v = 0; v < 8; ++v) { const float x = acc[ct][v]; s += x; q += x * x; }
      atomicAdd(&sSum[ct * 16 + l], s);
      atomicAdd(&sSq[ct * 16 + l], q);
    }
    __syncthreads();
    if (tid < OUTC) { atomicAdd(&gsum[tid], sSum[tid]); atomicAdd(&gsq[tid], sSq[tid]); }
    return;
  }

  // phase 3: BN2 + ReLU + max over the 16 neighbor rows of this tile.
#pragma unroll
  for (int ct = 0; ct < 8; ++ct) {
    const int ch = ct * 16 + l;
    const float sa = pa2[ch];
    const float sb = pb2[ch];
    float m = 0.f;   // ReLU outputs are >= 0
#pragma unroll
    for (int v = 0; v < 8; ++v) {
      const float x = fmaxf(acc[ct][v] * sa + sb, 0.f);
      m = fmaxf(m, x);
    }
    const float othr = __shfl_xor(m, 16, 32);
    m = fmaxf(m, othr);
    if (hh == 0) {
      const int pidx = blockIdx.x * 8 + wv;   // b*N + n of this tile
      out[(size_t)pidx * OUTC + ch] = m;
    }
  }
}

// ---------------------------------------------------------------------------
// Finalize BN params: a = gamma * rsqrt(var+eps), b = beta - mean*a
// ---------------------------------------------------------------------------
__global__ void finalize_bn(const float* __restrict__ sum, const float* __restrict__ sq,
                            const float* __restrict__ gamma, const float* __restrict__ beta,
                            float* __restrict__ pa, float* __restrict__ pb, float inv_count) {
  const int o = threadIdx.x;
  const float mean = sum[o] * inv_count;
  const float var  = sq[o] * inv_count - mean * mean;
  const float s = gamma[o] * rsqrtf(var + 1e-5f);
  pa[o] = s;
  pb[o] = beta[o] - mean * s;
}

extern "C" void kernel_launch(void* const* d_in, const int* in_sizes, int n_in,
                              void* d_out, int out_size, void* d_ws, size_t ws_size,
                              hipStream_t stream) {
  (void)in_sizes; (void)n_in; (void)out_size; (void)ws_size;
  const float* feats = (const float*)d_in[0];
  const float* xyz   = (const float*)d_in[1];
  const float* w1    = (const float*)d_in[2];
  const float* g1    = (const float*)d_in[3];
  const float* be1   = (const float*)d_in[4];
  const float* w2    = (const float*)d_in[5];
  const float* g2    = (const float*)d_in[6];
  const float* be2   = (const float*)d_in[7];
  float* out = (float*)d_out;

  int*   idxbuf = (int*)d_ws;
  float* stats  = (float*)((char*)d_ws + (size_t)MTOT * sizeof(int));
  float* sum1 = stats +   0; float* sq1 = stats + 128;
  float* sum2 = stats + 256; float* sq2 = stats + 384;
  float* pa1  = stats + 512; float* pb1 = stats + 640;
  float* pa2  = stats + 768; float* pb2 = stats + 896;

  hipMemsetAsync(stats, 0, 512 * sizeof(float), stream);

  knn_kernel<<<64, 256, 0, stream>>>(xyz, idxbuf);

  const int blocks = MTOT / 128;  // 2048
  const float invM = 1.0f / (float)MTOT;

  fused_kernel<<<blocks, 256, 0, stream>>>(feats, idxbuf, w1, w2,
                                           pa1, pb1, pa2, pb2,
                                           sum1, sq1, out, 1);
  finalize_bn<<<1, 128, 0, stream>>>(sum1, sq1, g1, be1, pa1, pb1, invM);
  fused_kernel<<<blocks, 256, 0, stream>>>(feats, idxbuf, w1, w2,
                                           pa1, pb1, pa2, pb2,
                                           sum2, sq2, out, 2);
  finalize_bn<<<1, 128, 0, stream>>>(sum2, sq2, g2, be2, pa2, pb2, invM);
  fused_kernel<<<blocks, 256, 0, stream>>>(feats, idxbuf, w1, w2,
                                           pa1, pb1, pa2, pb2,
                                           sum1, sq1, out, 3);
}